// gen_U0_5815385718888
// MI455X (gfx1250) — compile-verified
//
#include <hip/hip_runtime.h>
#include <hip/hip_bf16.h>

#define OUT_RES 20
#define NN      48
#define K2      6
#define EYD     120
#define ENN_OUT 14400
#define E0_HW   1060
#define NODES   128
#define NSAMP   (NN * NN)          // 2304

typedef float v2f __attribute__((ext_vector_type(2)));
typedef float v8f __attribute__((ext_vector_type(8)));

static __device__ __forceinline__ v8f splat8(float x) {
    v8f r;
#pragma unroll
    for (int i = 0; i < 8; ++i) r[i] = x;
    return r;
}

// ---------------------------------------------------------------------------
// Kernel A: both MLP trunks through layer 2.
//   neff[p] = relu(relu(x nW0+b) nW1+b) nW2 + nb2          -> ws
//   g2[p,:] = relu(relu(x eW0+b) eW1 + eb1)                -> ws
// One block (128 threads) per sample; W1 matrices (64KB each) stay hot in L2.
// ---------------------------------------------------------------------------
__global__ __launch_bounds__(NODES)
void mlp_hidden_kernel(const float* __restrict__ hs,
                       const float* __restrict__ nW0, const float* __restrict__ nb0,
                       const float* __restrict__ nW1, const float* __restrict__ nb1,
                       const float* __restrict__ nW2, const float* __restrict__ nb2,
                       const float* __restrict__ eW0, const float* __restrict__ eb0,
                       const float* __restrict__ eW1, const float* __restrict__ eb1,
                       float* __restrict__ g2, float* __restrict__ neff_out) {
    __shared__ float h1s[NODES];
    __shared__ float g1s[NODES];
    __shared__ float red[NODES];
    const int p = blockIdx.x;
    const int k = threadIdx.x;
    const float t = hs[p];
    h1s[k] = fmaxf(fmaf(t, nW0[k], nb0[k]), 0.f);
    g1s[k] = fmaxf(fmaf(t, eW0[k], eb0[k]), 0.f);
    __syncthreads();
    float an = nb1[k];
    float ae = eb1[k];
#pragma unroll 4
    for (int j = 0; j < NODES; ++j) {
        an = fmaf(h1s[j], nW1[j * NODES + k], an);
        ae = fmaf(g1s[j], eW1[j * NODES + k], ae);
    }
    g2[p * NODES + k] = fmaxf(ae, 0.f);      // enn hidden-2 (A operand of big GEMM)
    red[k] = fmaxf(an, 0.f) * nW2[k];        // nW2 is [128,1]
    __syncthreads();
#pragma unroll
    for (int s = NODES / 2; s > 0; s >>= 1) {
        if (k < s) red[k] += red[k + s];
        __syncthreads();
    }
    if (k == 0) neff_out[p] = red[0] + nb2[0];
}

// ---------------------------------------------------------------------------
// Kernel B: Ey = g2 [2304x128] @ eW2 [128x14400] + eb2, exact fp32 via
// V_WMMA_F32_16X16X4_F32. One wave per block; each wave owns a 64x64 output
// tile: 4 M-tiles x 4 N-tiles = 16 v8f accumulators. Each B float2 feeds 4
// WMMAs (4x less L2 B-traffic than a 16-row tile); A is a b64 load per M-tile.
//   A layout : lane&15 = row, half(lane>=16) selects K pair {k,k+1 | k+2,k+3}
//   B layout : lane&15 = column, half selects K pair
//   C layout : vgpr v -> row v + 8*half, lane&15 = column
// ---------------------------------------------------------------------------
__global__ __launch_bounds__(32)
void ey_gemm_wmma(const float* __restrict__ A,     // g2   [2304][128]
                  const float* __restrict__ B,     // eW2  [128][14400]
                  const float* __restrict__ bias,  // eb2  [14400]
                  float* __restrict__ Ey) {        // out  [2304][14400]
    const int n0   = blockIdx.x * 64;
    const int m0   = blockIdx.y * 64;
    const int lane = threadIdx.x;
    const int sub  = lane & 15;
    const int half = lane >> 4;

    v8f c[4][4];                                   // [m-tile][n-tile]
#pragma unroll
    for (int ct = 0; ct < 4; ++ct) {
        const v8f bv = splat8(bias[n0 + ct * 16 + sub]);
#pragma unroll
        for (int mt = 0; mt < 4; ++mt) c[mt][ct] = bv;
    }

    const float* arow = A + (size_t)(m0 + sub) * NODES;

    for (int kk = 0; kk < NODES; kk += 4) {
        const int ka = kk + 2 * half;
        v2f a[4];
#pragma unroll
        for (int mt = 0; mt < 4; ++mt)
            a[mt] = *(const v2f*)(arow + (size_t)mt * 16 * NODES + ka);

        const float* Bp = B + (size_t)ka * ENN_OUT + n0 + sub;
        // speculative prefetch of next K-step's B rows (global_prefetch_b8)
        __builtin_prefetch(Bp + 4 * ENN_OUT, 0, 1);
#pragma unroll
        for (int ct = 0; ct < 4; ++ct) {
            v2f b;
            b.x = Bp[ct * 16];
            b.y = Bp[ct * 16 + ENN_OUT];
#pragma unroll
            for (int mt = 0; mt < 4; ++mt)
                c[mt][ct] = __builtin_amdgcn_wmma_f32_16x16x4_f32(
                    false, a[mt], false, b, (short)0, c[mt][ct], false, false);
        }
    }

#pragma unroll
    for (int mt = 0; mt < 4; ++mt) {
#pragma unroll
        for (int v = 0; v < 8; ++v) {
            const int m = m0 + mt * 16 + v + 8 * half;
            float* orow = Ey + (size_t)m * ENN_OUT + n0 + sub;
#pragma unroll
            for (int ct = 0; ct < 4; ++ct)
                orow[ct * 16] = c[mt][ct][v];
        }
    }
}

// ---------------------------------------------------------------------------
// Kernel C: T[p] = 0.005 * neff/(neff+1) * sum_e Ey[p,e] * (E0r+i*E0i)[gather]
// E0 (~9MB complex) is L2-resident; Ey is streamed once.
// Gather math reproduces the flat row-major reshape of the reference exactly:
//   e -> (i,a,j,b); idx = p*400 + a*20 + b; rr = idx/960; cc = idx%960;
//   E0 element = E0[i*20+rr][j*20+cc]
// ---------------------------------------------------------------------------
__global__ __launch_bounds__(256)
void tsum_kernel(const float* __restrict__ Ey,
                 const float* __restrict__ E0r,
                 const float* __restrict__ E0i,
                 const float* __restrict__ neff,
                 float* __restrict__ Tout) {
    const int p   = blockIdx.x;
    const int tid = threadIdx.x;
    float ar = 0.f, ai = 0.f;
    const float* ey = Ey + (size_t)p * ENN_OUT;
    for (int e = tid; e < ENN_OUT; e += 256) {
        const int y = e / EYD;
        const int z = e - y * EYD;
        const int i = y / OUT_RES;
        const int a = y - i * OUT_RES;
        const int j = z / OUT_RES;
        const int b = z - j * OUT_RES;
        const int idx = p * 400 + a * OUT_RES + b;
        const int rr  = idx / 960;
        const int cc  = idx - rr * 960;
        const int adr = (i * OUT_RES + rr) * E0_HW + j * OUT_RES + cc;
        const float v = ey[e];
        ar = fmaf(v, E0r[adr], ar);
        ai = fmaf(v, E0i[adr], ai);
    }
    __shared__ float sr[256];
    __shared__ float si[256];
    sr[tid] = ar;
    si[tid] = ai;
    __syncthreads();
#pragma unroll
    for (int s = 128; s > 0; s >>= 1) {
        if (tid < s) { sr[tid] += sr[tid + s]; si[tid] += si[tid + s]; }
        __syncthreads();
    }
    if (tid == 0) {
        const float nf    = neff[p];
        const float eta   = nf / (nf + 1.0f);          // N0 = 1
        const float scale = 0.005f * eta;              // 2*C_EPSILON*dx*dx, dx=0.05
        Tout[2 * p]     = scale * sr[0];
        Tout[2 * p + 1] = scale * si[0];
    }
}

extern "C" void kernel_launch(void* const* d_in, const int* in_sizes, int n_in,
                              void* d_out, int out_size, void* d_ws, size_t ws_size,
                              hipStream_t stream) {
    (void)in_sizes; (void)n_in; (void)out_size; (void)ws_size;
    const float* hs  = (const float*)d_in[0];
    const float* E0r = (const float*)d_in[1];
    const float* E0i = (const float*)d_in[2];
    const float* nW0 = (const float*)d_in[3];
    const float* nb0 = (const float*)d_in[4];
    const float* nW1 = (const float*)d_in[5];
    const float* nb1 = (const float*)d_in[6];
    const float* nW2 = (const float*)d_in[7];
    const float* nb2 = (const float*)d_in[8];
    const float* eW0 = (const float*)d_in[9];
    const float* eb0 = (const float*)d_in[10];
    const float* eW1 = (const float*)d_in[11];
    const float* eb1 = (const float*)d_in[12];
    const float* eW2 = (const float*)d_in[13];
    const float* eb2 = (const float*)d_in[14];

    float* g2   = (float*)d_ws;                 // [2304][128]
    float* nf   = g2 + (size_t)NSAMP * NODES;   // [2304]
    float* Ey   = (float*)d_out;                // [2304][14400]
    float* Tout = Ey + (size_t)NSAMP * ENN_OUT; // [2304] complex interleaved

    mlp_hidden_kernel<<<NSAMP, NODES, 0, stream>>>(hs, nW0, nb0, nW1, nb1, nW2, nb2,
                                                   eW0, eb0, eW1, eb1, g2, nf);

    dim3 gridB(ENN_OUT / 64, NSAMP / 64);       // (225, 36)
    ey_gemm_wmma<<<gridB, 32, 0, stream>>>(g2, eW2, eb2, Ey);

    tsum_kernel<<<NSAMP, 256, 0, stream>>>(Ey, E0r, E0i, nf, Tout);
}